// GAT_66838281060947
// MI455X (gfx1250) — compile-verified
//
#include <hip/hip_runtime.h>
#include <hip/hip_bf16.h>

// ---------------------------------------------------------------------------
// GATv2 (3 layers) for MI455X / gfx1250.
//  - Dense transforms via V_WMMA_F32_16X16X4_F32 (exact fp32 matrix pipe).
//  - Edge phase: one wave32 per (edge, head) -> contiguous 256B row reads,
//    L2-resident gathers; scatter via GLOBAL_ATOMIC_ADD_F32.
// ---------------------------------------------------------------------------

typedef float v2f __attribute__((ext_vector_type(2)));
typedef float v8f __attribute__((ext_vector_type(8)));

#define NEG_SLOPE 0.2f

// ---------------- WMMA dual GEMM: XL = X@Wl, XR = X@Wr ---------------------
// X: [N,K] row-major.  Wl/Wr: [K,M] row-major.  XL/XR: [N,M] row-major.
// One wave per 16x16 output tile; K stepped by 4 (V_WMMA_F32_16X16X4_F32).
// A 16x4 layout (ISA 7.12.2): lane m=lane&15, VGPR0/1 hold K = (lane>>4)*2 +0/1
// B 4x16 layout: lane n=lane&15, VGPR0/1 hold rows K = (lane>>4)*2 +0/1
// C/D 16x16: VGPR r -> row r + 8*(lane>>4), col lane&15
__global__ void gemm_dual_wmma(const float* __restrict__ X,
                               const float* __restrict__ Wl,
                               const float* __restrict__ Wr,
                               float* __restrict__ XL,
                               float* __restrict__ XR,
                               int K, int M) {
  const int lane    = threadIdx.x;                    // 0..31
  const int rowTile = blockIdx.x;                     // N/16 tiles
  const int colTile = blockIdx.y * 4 + threadIdx.y;   // M/16 tiles
  const int m     = lane & 15;
  const int khalf = (lane >> 4) * 2;                  // 0 or 2
  const int row   = rowTile * 16 + m;                 // A row for this lane
  const int col   = colTile * 16 + m;                 // B col for this lane

  v8f accL = {};
  v8f accR = {};
  const float* xrow = X + (size_t)row * K;

  for (int kk = 0; kk < K; kk += 4) {
    const int k0 = kk + khalf;
    v2f a;
    a.x = xrow[k0];
    a.y = xrow[k0 + 1];
    v2f bl, br;
    bl.x = Wl[(size_t)k0 * M + col];
    bl.y = Wl[(size_t)(k0 + 1) * M + col];
    br.x = Wr[(size_t)k0 * M + col];
    br.y = Wr[(size_t)(k0 + 1) * M + col];
    accL = __builtin_amdgcn_wmma_f32_16x16x4_f32(false, a, false, bl,
                                                 (short)0, accL, false, false);
    accR = __builtin_amdgcn_wmma_f32_16x16x4_f32(false, a, false, br,
                                                 (short)0, accR, false, false);
  }

  const int rbase = (lane >> 4) * 8;
  const int ccol  = colTile * 16 + (lane & 15);
  for (int r = 0; r < 8; ++r) {
    const size_t o = (size_t)(rowTile * 16 + rbase + r) * M + ccol;
    XL[o] = accL[r];
    XR[o] = accR[r];
  }
}

// ------------------- float <-> order-preserving uint -----------------------
__device__ __forceinline__ unsigned f2ord(float f) {
  unsigned u = __float_as_uint(f);
  return (u & 0x80000000u) ? ~u : (u | 0x80000000u);
}
__device__ __forceinline__ float ord2f(unsigned u) {
  return (u & 0x80000000u) ? __uint_as_float(u & 0x7fffffffu)
                           : __uint_as_float(~u);
}

// ---------------- edge scores: e[edge,h] = att_h . lrelu(xl[src]+xr[dst]) --
// one wave per (edge, head); 32 lanes x 2 floats = C=64 channels
__global__ void edge_scores(const float* __restrict__ XL,
                            const float* __restrict__ XR,
                            const int* __restrict__ ei,
                            const float* __restrict__ att,
                            float* __restrict__ esc,
                            int Eg, int Et, int H, int M) {
  const int gtid = blockIdx.x * blockDim.x + threadIdx.x;
  const int w = gtid >> 5, lane = gtid & 31;
  if (w >= Et * H) return;
  const int e = w / H, h = w - e * H;
  int src, dst;
  if (e < Eg) { src = ei[e]; dst = ei[Eg + e]; }
  else        { src = dst = e - Eg; }

  const int c = lane * 2;
  const float2 xl = *(const float2*)(XL + (size_t)src * M + h * 64 + c);
  const float2 xr = *(const float2*)(XR + (size_t)dst * M + h * 64 + c);
  const float2 at = *(const float2*)(att + h * 64 + c);
  float mx = xl.x + xr.x, my = xl.y + xr.y;
  mx = mx > 0.f ? mx : NEG_SLOPE * mx;
  my = my > 0.f ? my : NEG_SLOPE * my;
  float s = at.x * mx + at.y * my;
  for (int off = 16; off > 0; off >>= 1) s += __shfl_xor(s, off, 32);
  if (lane == 0) esc[(size_t)e * H + h] = s;
}

// ---------------- segment max over dst -------------------------------------
__global__ void seg_max(const float* __restrict__ esc,
                        const int* __restrict__ ei,
                        unsigned* __restrict__ nmax,
                        int Eg, int Et, int H) {
  const int t = blockIdx.x * blockDim.x + threadIdx.x;
  if (t >= Et * H) return;
  const int e = t / H, h = t - e * H;
  const int dst = (e < Eg) ? ei[Eg + e] : (e - Eg);
  atomicMax(&nmax[(size_t)dst * H + h], f2ord(esc[t]));
}

// ---------------- exp(e - max) and segment sum -----------------------------
__global__ void seg_expsum(float* __restrict__ esc,
                           const int* __restrict__ ei,
                           const unsigned* __restrict__ nmax,
                           float* __restrict__ denom,
                           int Eg, int Et, int H) {
  const int t = blockIdx.x * blockDim.x + threadIdx.x;
  if (t >= Et * H) return;
  const int e = t / H, h = t - e * H;
  const int dst = (e < Eg) ? ei[Eg + e] : (e - Eg);
  const float mx = ord2f(nmax[(size_t)dst * H + h]);
  const float ex = __expf(esc[t] - mx);
  esc[t] = ex;
  atomicAdd(&denom[(size_t)dst * H + h], ex);
}

// ---------------- aggregate: acc[dst] += alpha * xl[src] -------------------
// one wave per (edge, head); 2 channels per lane
__global__ void aggregate(const float* __restrict__ XL,
                          float* __restrict__ acc,
                          const float* __restrict__ esc,
                          const float* __restrict__ denom,
                          const int* __restrict__ ei,
                          int Eg, int Et, int H, int M) {
  const int gtid = blockIdx.x * blockDim.x + threadIdx.x;
  const int w = gtid >> 5, lane = gtid & 31;
  if (w >= Et * H) return;
  const int e = w / H, h = w - e * H;
  int src, dst;
  if (e < Eg) { src = ei[e]; dst = ei[Eg + e]; }
  else        { src = dst = e - Eg; }
  const float alpha = esc[(size_t)e * H + h] / denom[(size_t)dst * H + h];
  const int c = lane * 2;
  const float2 xl = *(const float2*)(XL + (size_t)src * M + h * 64 + c);
  float* o = acc + (size_t)dst * M + h * 64 + c;
  atomicAdd(o,     alpha * xl.x);
  atomicAdd(o + 1, alpha * xl.y);
}

// ---------------- bias + optional ELU --------------------------------------
__global__ void bias_act(const float* __restrict__ acc,
                         const float* __restrict__ bias,
                         float* __restrict__ out,
                         int M, int do_elu, long long total) {
  const long long t = (long long)blockIdx.x * blockDim.x + threadIdx.x;
  if (t >= total) return;
  const int j = (int)(t % M);
  float v = acc[t] + bias[j];
  if (do_elu) v = v > 0.f ? v : expm1f(v);
  out[t] = v;
}

// ---------------- fill -----------------------------------------------------
__global__ void fill_u32(unsigned* __restrict__ p, unsigned v, long long n) {
  const long long t = (long long)blockIdx.x * blockDim.x + threadIdx.x;
  if (t < n) p[t] = v;
}

// ---------------------------------------------------------------------------
extern "C" void kernel_launch(void* const* d_in, const int* in_sizes, int n_in,
                              void* d_out, int out_size, void* d_ws, size_t ws_size,
                              hipStream_t stream) {
  const int N  = 50000;
  const int Eg = 800000;
  const int Et = Eg + N;  // with self-loops

  const float* x  = (const float*)d_in[0];
  const int*   ei = (const int*)d_in[1];

  struct Layer { const float *Wl, *Wr, *att, *bias; int K, H, M; };
  Layer L[3] = {
    { (const float*)d_in[2],  (const float*)d_in[3],
      (const float*)d_in[4],  (const float*)d_in[5],  128, 4, 256 },
    { (const float*)d_in[6],  (const float*)d_in[7],
      (const float*)d_in[8],  (const float*)d_in[9],  256, 4, 256 },
    { (const float*)d_in[10], (const float*)d_in[11],
      (const float*)d_in[12], (const float*)d_in[13], 256, 1, 64  },
  };

  // workspace carve-up (fp32 words)
  float* ws    = (float*)d_ws;
  float* xl    = ws;                       // N*256
  float* xr    = xl + (size_t)N * 256;     // N*256
  float* esc   = xr + (size_t)N * 256;     // Et*4
  unsigned* nmax = (unsigned*)(esc + (size_t)Et * 4);  // N*4
  float* denom = (float*)(nmax + (size_t)N * 4);       // N*4
  float* acc   = denom + (size_t)N * 4;    // N*256
  float* actbuf = acc + (size_t)N * 256;   // N*256

  const float* act = x;

  for (int li = 0; li < 3; ++li) {
    const int K = L[li].K, H = L[li].H, M = L[li].M;

    // dense transforms (WMMA fp32)
    dim3 gg(N / 16, M / 64);
    dim3 gb(32, 4);
    gemm_dual_wmma<<<gg, gb, 0, stream>>>(act, L[li].Wl, L[li].Wr, xl, xr, K, M);

    // init reductions
    {
      long long nH = (long long)N * H;
      long long nM = (long long)N * M;
      fill_u32<<<(int)((nH + 255) / 256), 256, 0, stream>>>(nmax, 0u, nH);
      fill_u32<<<(int)((nH + 255) / 256), 256, 0, stream>>>((unsigned*)denom, 0u, nH);
      fill_u32<<<(int)((nM + 255) / 256), 256, 0, stream>>>((unsigned*)acc, 0u, nM);
    }

    const long long nWaves = (long long)Et * H;
    const int waveBlocks = (int)((nWaves + 7) / 8);        // 8 waves / 256-thr block
    const int thrBlocks  = (int)((nWaves + 255) / 256);    // thread-per-(e,h)

    edge_scores<<<waveBlocks, 256, 0, stream>>>(xl, xr, ei, L[li].att, esc, Eg, Et, H, M);
    seg_max    <<<thrBlocks, 256, 0, stream>>>(esc, ei, nmax, Eg, Et, H);
    seg_expsum <<<thrBlocks, 256, 0, stream>>>(esc, ei, nmax, denom, Eg, Et, H);
    aggregate  <<<waveBlocks, 256, 0, stream>>>(xl, acc, esc, denom, ei, Eg, Et, H, M);

    const long long total = (long long)N * M;
    float* dst = (li < 2) ? actbuf : (float*)d_out;
    bias_act<<<(int)((total + 255) / 256), 256, 0, stream>>>(
        acc, L[li].bias, dst, M, (li < 2) ? 1 : 0, total);

    act = actbuf;
  }
}